// SpotGuidedGeoTransformer_74388833566807
// MI455X (gfx1250) — compile-verified
//
#include <hip/hip_runtime.h>

typedef float v2f __attribute__((ext_vector_type(2)));
typedef float v8f __attribute__((ext_vector_type(8)));

#define NFINE   4096
#define NCOARSE 512
#define CFDIM   256
#define CCDIM   1024
#define HDIM    256
#define SEEDN   128
#define SPOTK   16

// ---------------------------------------------------------------------------
// WMMA f32 16x16x4 GEMM.  C[M x N] = A[M x K] * op(B) (+ bias per column).
//   OPB==0 : B is row-major [K x N]            (projections: feats @ W)
//   OPB==1 : B is row-major [N x K], compute A*B^T  (scores: hf_ref @ hf_src^T)
// One wave computes a 16x64 strip (4 accumulators, A-fragment reused 4x).
// Requires M%16==0, N%64==0, K%4==0.  Block = 256 threads = 8 waves.
// ---------------------------------------------------------------------------
template <int OPB, bool BIAS>
__global__ void wmma_gemm_f32(const float* __restrict__ A,
                              const float* __restrict__ B,
                              const float* __restrict__ bias,
                              float* __restrict__ C,
                              int M, int N, int K) {
  const int lane = threadIdx.x & 31;
  const int wave = threadIdx.x >> 5;
  const int half = lane >> 4;      // 0: lanes 0-15, 1: lanes 16-31
  const int r    = lane & 15;

  const int strips_n = N >> 6;                         // 64-wide strips
  const int strip    = blockIdx.x * 8 + wave;
  const int total    = (M >> 4) * strips_n;
  if (strip >= total) return;
  const int tm    = strip / strips_n;
  const int sn    = strip % strips_n;
  const int mrow  = tm * 16 + r;                       // A row this lane feeds
  const int ncol0 = sn * 64 + r;                       // first B column

  v8f acc0 = {}; v8f acc1 = {}; v8f acc2 = {}; v8f acc3 = {};
  const float* Arow = A + (size_t)mrow * K;

#pragma unroll 4
  for (int k0 = 0; k0 < K; k0 += 4) {
    const int ka = k0 + 2 * half;
    v2f a = *(const v2f*)(Arow + ka);
    v2f b0, b1, b2, b3;
    if (OPB) {
      b0 = *(const v2f*)(B + (size_t)(ncol0)      * K + ka);
      b1 = *(const v2f*)(B + (size_t)(ncol0 + 16) * K + ka);
      b2 = *(const v2f*)(B + (size_t)(ncol0 + 32) * K + ka);
      b3 = *(const v2f*)(B + (size_t)(ncol0 + 48) * K + ka);
    } else {
      const float* Bk0 = B + (size_t)ka * N;
      const float* Bk1 = B + (size_t)(ka + 1) * N;
      b0.x = Bk0[ncol0];      b0.y = Bk1[ncol0];
      b1.x = Bk0[ncol0 + 16]; b1.y = Bk1[ncol0 + 16];
      b2.x = Bk0[ncol0 + 32]; b2.y = Bk1[ncol0 + 32];
      b3.x = Bk0[ncol0 + 48]; b3.y = Bk1[ncol0 + 48];
    }
    acc0 = __builtin_amdgcn_wmma_f32_16x16x4_f32(false, a, false, b0, (short)0, acc0, false, false);
    acc1 = __builtin_amdgcn_wmma_f32_16x16x4_f32(false, a, false, b1, (short)0, acc1, false, false);
    acc2 = __builtin_amdgcn_wmma_f32_16x16x4_f32(false, a, false, b2, (short)0, acc2, false, false);
    acc3 = __builtin_amdgcn_wmma_f32_16x16x4_f32(false, a, false, b3, (short)0, acc3, false, false);
  }

  float bb0 = 0.f, bb1 = 0.f, bb2 = 0.f, bb3 = 0.f;
  if (BIAS) {
    bb0 = bias[ncol0]; bb1 = bias[ncol0 + 16];
    bb2 = bias[ncol0 + 32]; bb3 = bias[ncol0 + 48];
  }
#pragma unroll
  for (int v = 0; v < 8; ++v) {
    const int rowi = tm * 16 + v + 8 * half;           // C layout: M = v + 8*half
    float* Crow = C + (size_t)rowi * N;
    Crow[ncol0]      = acc0[v] + bb0;
    Crow[ncol0 + 16] = acc1[v] + bb1;
    Crow[ncol0 + 32] = acc2[v] + bb2;
    Crow[ncol0 + 48] = acc3[v] + bb3;
  }
}

// ---------------------------------------------------------------------------
// Row softmax statistics: one block (256 thr) per row.
// ---------------------------------------------------------------------------
__global__ void row_stats_kernel(const float* __restrict__ S,
                                 float* __restrict__ rmax, float* __restrict__ rsum,
                                 int Ncols) {
  __shared__ float sh[256];
  const int i = blockIdx.x, tid = threadIdx.x;
  const float* row = S + (size_t)i * Ncols;
  float mx = -3.0e38f;
  for (int c = tid; c < Ncols; c += 256) mx = fmaxf(mx, row[c]);
  sh[tid] = mx; __syncthreads();
  for (int s = 128; s > 0; s >>= 1) { if (tid < s) sh[tid] = fmaxf(sh[tid], sh[tid + s]); __syncthreads(); }
  mx = sh[0]; __syncthreads();
  float sm = 0.f;
  for (int c = tid; c < Ncols; c += 256) sm += expf(row[c] - mx);
  sh[tid] = sm; __syncthreads();
  for (int s = 128; s > 0; s >>= 1) { if (tid < s) sh[tid] += sh[tid + s]; __syncthreads(); }
  if (tid == 0) { rmax[i] = mx; rsum[i] = sh[0]; }
}

// Column softmax statistics: one thread per column, online max/sum (coalesced).
__global__ void col_stats_kernel(const float* __restrict__ S,
                                 float* __restrict__ cmax, float* __restrict__ csum,
                                 int Nrows, int Ncols) {
  const int c = blockIdx.x * 256 + threadIdx.x;
  if (c >= Ncols) return;
  float mx = -3.0e38f, sm = 0.f;
  for (int r = 0; r < Nrows; ++r) {
    float v = S[(size_t)r * Ncols + c];
    if (v > mx) { sm = sm * expf(mx - v) + 1.f; mx = v; }
    else        { sm += expf(v - mx); }
  }
  cmax[c] = mx; csum[c] = sm;
}

// Dual-softmax rewrite in place + per-row max/argmax (first index wins ties).
__global__ void dual_row_kernel(float* __restrict__ S,
                                const float* __restrict__ rmax, const float* __restrict__ rsum,
                                const float* __restrict__ cmax, const float* __restrict__ csum,
                                float* __restrict__ conf, int* __restrict__ mi, int Ncols) {
  __shared__ float sv[256];
  __shared__ int   si[256];
  const int i = blockIdx.x, tid = threadIdx.x;
  const float rm = rmax[i];
  const float ri = 1.0f / rsum[i];
  float* row = S + (size_t)i * Ncols;
  float bv = -3.0e38f; int bi = 0;
  for (int c = tid; c < Ncols; c += 256) {
    float s = row[c];
    float p = expf(s - rm) * ri * (expf(s - cmax[c]) / csum[c]);
    row[c] = p;
    if (p > bv) { bv = p; bi = c; }
  }
  sv[tid] = bv; si[tid] = bi; __syncthreads();
  for (int s = 128; s > 0; s >>= 1) {
    if (tid < s) {
      if (sv[tid + s] > sv[tid] || (sv[tid + s] == sv[tid] && si[tid + s] < si[tid])) {
        sv[tid] = sv[tid + s]; si[tid] = si[tid + s];
      }
    }
    __syncthreads();
  }
  if (tid == 0) { conf[i] = sv[0]; mi[i] = si[0]; }
}

// Per-column max/argmax of final scores (first row wins ties).
__global__ void col_argmax_kernel(const float* __restrict__ S,
                                  float* __restrict__ conf, int* __restrict__ mi,
                                  int Nrows, int Ncols) {
  const int c = blockIdx.x * 256 + threadIdx.x;
  if (c >= Ncols) return;
  float bv = -3.0e38f; int bi = 0;
  for (int r = 0; r < Nrows; ++r) {
    float v = S[(size_t)r * Ncols + c];
    if (v > bv) { bv = v; bi = r; }
  }
  conf[c] = bv; mi[c] = bi;
}

// q[i] = pts[idx[i]]  (3 floats)
__global__ void gather_pts_kernel(const float* __restrict__ pts, const int* __restrict__ idx,
                                  float* __restrict__ q, int n) {
  const int i = blockIdx.x * 256 + threadIdx.x;
  if (i >= n) return;
  const int m = idx[i];
  q[i * 3 + 0] = pts[m * 3 + 0];
  q[i * 3 + 1] = pts[m * 3 + 1];
  q[i * 3 + 2] = pts[m * 3 + 2];
}

// compat[i] = mean_j relu(1 - |d(p_i,p_j) - d(q_i,q_j)| / 3)   (points tiled in LDS)
__global__ void compat_kernel(const float* __restrict__ P, const float* __restrict__ Q,
                              float* __restrict__ out, int n) {
  __shared__ float sp[256 * 3];
  __shared__ float sq[256 * 3];
  const int i = blockIdx.x * 256 + threadIdx.x;
  const float px = P[i * 3], py = P[i * 3 + 1], pz = P[i * 3 + 2];
  const float qx = Q[i * 3], qy = Q[i * 3 + 1], qz = Q[i * 3 + 2];
  float acc = 0.f;
  for (int j0 = 0; j0 < n; j0 += 256) {
    for (int t = threadIdx.x; t < 256 * 3; t += 256) {
      sp[t] = P[j0 * 3 + t];
      sq[t] = Q[j0 * 3 + t];
    }
    __syncthreads();
    for (int jj = 0; jj < 256; ++jj) {
      float dx = px - sp[jj * 3], dy = py - sp[jj * 3 + 1], dz = pz - sp[jj * 3 + 2];
      float dref = dx * dx + dy * dy + dz * dz;
      float ex = qx - sq[jj * 3], ey = qy - sq[jj * 3 + 1], ez = qz - sq[jj * 3 + 2];
      float dsrc = ex * ex + ey * ey + ez * ez;
      float v = 1.f - fabsf(dref - dsrc) * (1.f / 3.0f);
      acc += (v > 0.f) ? v : 0.f;
    }
    __syncthreads();
  }
  out[i] = acc * (1.0f / (float)n);
}

// ---------------------------------------------------------------------------
// KNN: per query point, KSEL nearest of nP database points (on-the-fly dists,
// database tiled through LDS, sorted register insertion; ties keep low index).
// Writes int indices (outI) or float indices (outF).
// ---------------------------------------------------------------------------
template <int KSEL>
__global__ void knn_kernel(const float* __restrict__ Qp, int nQ,
                           const float* __restrict__ Pp, int nP,
                           int* __restrict__ outI, float* __restrict__ outF) {
  __shared__ float sp[256 * 3];
  const int i = blockIdx.x * 256 + threadIdx.x;
  float qx = 0.f, qy = 0.f, qz = 0.f;
  if (i < nQ) { qx = Qp[i * 3]; qy = Qp[i * 3 + 1]; qz = Qp[i * 3 + 2]; }
  float bd[KSEL]; int bi[KSEL];
#pragma unroll
  for (int k = 0; k < KSEL; ++k) { bd[k] = 3.0e38f; bi[k] = 0; }
  for (int j0 = 0; j0 < nP; j0 += 256) {
    for (int t = threadIdx.x; t < 256 * 3; t += 256) sp[t] = Pp[j0 * 3 + t];
    __syncthreads();
    for (int jj = 0; jj < 256; ++jj) {
      float dx = qx - sp[jj * 3], dy = qy - sp[jj * 3 + 1], dz = qz - sp[jj * 3 + 2];
      float d = dx * dx + dy * dy + dz * dz;
      if (d < bd[KSEL - 1]) {
        bd[KSEL - 1] = d; bi[KSEL - 1] = j0 + jj;
#pragma unroll
        for (int t = KSEL - 1; t > 0; --t) {
          if (bd[t] < bd[t - 1]) {
            float td = bd[t]; bd[t] = bd[t - 1]; bd[t - 1] = td;
            int   ti = bi[t]; bi[t] = bi[t - 1]; bi[t - 1] = ti;
          }
        }
      }
    }
    __syncthreads();
  }
  if (i < nQ) {
#pragma unroll
    for (int k = 0; k < KSEL; ++k) {
      if (outI) outI[i * KSEL + k] = bi[k];
      else      outF[i * KSEL + k] = (float)bi[k];
    }
  }
}

// spot[i][k] = idx16[mi[i]][k], written as float
__global__ void spot_kernel(const int* __restrict__ mi, const int* __restrict__ idx16,
                            float* __restrict__ out, int n) {
  const int t = blockIdx.x * 256 + threadIdx.x;
  if (t >= n * SPOTK) return;
  const int i = t >> 4, k = t & 15;
  out[t] = (float)idx16[mi[i] * SPOTK + k];
}

// Faithful seeding: sel = compat < max(compat)*0.5;  w = sel * conf * compat;
// top-128 indices of w with stable (ascending-index) tie-break.
__global__ void seed_kernel(const float* __restrict__ compat, const float* __restrict__ conf,
                            float* __restrict__ outSeed, int n) {
  __shared__ float w[NFINE];
  __shared__ float sv[256];
  __shared__ int   si[256];
  const int tid = threadIdx.x;
  float mx = -3.0e38f;
  for (int i = tid; i < n; i += 256) mx = fmaxf(mx, compat[i]);
  sv[tid] = mx; __syncthreads();
  for (int s = 128; s > 0; s >>= 1) { if (tid < s) sv[tid] = fmaxf(sv[tid], sv[tid + s]); __syncthreads(); }
  const float thr = sv[0] * 0.5f; __syncthreads();
  for (int i = tid; i < n; i += 256) {
    float c = compat[i];
    w[i] = (c < thr) ? conf[i] * c : 0.0f;
  }
  __syncthreads();
  for (int s = 0; s < SEEDN; ++s) {
    float bv = -3.0e38f; int bi = 0x7fffffff;
    for (int i = tid; i < n; i += 256) {
      float v = w[i];
      if (v > bv || (v == bv && i < bi)) { bv = v; bi = i; }
    }
    sv[tid] = bv; si[tid] = bi; __syncthreads();
    for (int st = 128; st > 0; st >>= 1) {
      if (tid < st) {
        if (sv[tid + st] > sv[tid] || (sv[tid + st] == sv[tid] && si[tid + st] < si[tid])) {
          sv[tid] = sv[tid + st]; si[tid] = si[tid + st];
        }
      }
      __syncthreads();
    }
    if (tid == 0) { outSeed[s] = (float)si[0]; w[si[0]] = -1.0f; }
    __syncthreads();
  }
}

// ---------------------------------------------------------------------------
extern "C" void kernel_launch(void* const* d_in, const int* in_sizes, int n_in,
                              void* d_out, int out_size, void* d_ws, size_t ws_size,
                              hipStream_t stream) {
  (void)in_sizes; (void)n_in; (void)out_size; (void)ws_size;
  const float* ref_points   = (const float*)d_in[0];
  const float* src_points   = (const float*)d_in[1];
  const float* ref_feats    = (const float*)d_in[2];
  const float* src_feats    = (const float*)d_in[3];
  const float* ref_points_c = (const float*)d_in[4];
  const float* src_points_c = (const float*)d_in[5];
  const float* ref_feats_c  = (const float*)d_in[6];
  const float* src_feats_c  = (const float*)d_in[7];
  const float* W1 = (const float*)d_in[8];
  const float* b1 = (const float*)d_in[9];
  const float* W2 = (const float*)d_in[10];
  const float* b2 = (const float*)d_in[11];
  float* out = (float*)d_out;

  // ---- output layout (return order, flat) ----
  size_t o = 0;
  float* o_scores  = out + o; o += (size_t)NFINE * NFINE;
  float* o_compatr = out + o; o += NFINE;
  float* o_compats = out + o; o += NFINE;
  float* o_seedr   = out + o; o += SEEDN;
  float* o_seeds   = out + o; o += SEEDN;
  float* o_spotr   = out + o; o += (size_t)NFINE * SPOTK;
  float* o_spots   = out + o; o += (size_t)NFINE * SPOTK;
  float* o_hcref   = out + o; o += (size_t)NCOARSE * HDIM;
  float* o_hcsrc   = out + o; o += (size_t)NCOARSE * HDIM;
  float* o_refup   = out + o; o += NFINE;
  float* o_srcup   = out + o; o += NFINE;
  float* o_refdn   = out + o; o += (size_t)NCOARSE * 8;
  float* o_srcdn   = out + o; o += (size_t)NCOARSE * 8;

  // ---- workspace layout ----
  char* w = (char*)d_ws;
  float* hf_ref = (float*)w; w += (size_t)NFINE * HDIM * 4;
  float* hf_src = (float*)w; w += (size_t)NFINE * HDIM * 4;
  float* rmax   = (float*)w; w += NFINE * 4;
  float* rsum   = (float*)w; w += NFINE * 4;
  float* cmax   = (float*)w; w += NFINE * 4;
  float* csum   = (float*)w; w += NFINE * 4;
  float* conf_r = (float*)w; w += NFINE * 4;
  float* conf_s = (float*)w; w += NFINE * 4;
  int*   mi_r   = (int*)w;   w += NFINE * 4;
  int*   mi_s   = (int*)w;   w += NFINE * 4;
  int*   ridx16 = (int*)w;   w += (size_t)NFINE * SPOTK * 4;
  int*   sidx16 = (int*)w;   w += (size_t)NFINE * SPOTK * 4;
  float* q_r    = (float*)w; w += NFINE * 3 * 4;
  float* q_s    = (float*)w; w += NFINE * 3 * 4;

  // ---- projections (WMMA f32) ----
  {
    int strips = (NFINE / 16) * (HDIM / 64);                  // 1024
    wmma_gemm_f32<0, true><<<(strips + 7) / 8, 256, 0, stream>>>(ref_feats, W2, b2, hf_ref, NFINE, HDIM, CFDIM);
    wmma_gemm_f32<0, true><<<(strips + 7) / 8, 256, 0, stream>>>(src_feats, W2, b2, hf_src, NFINE, HDIM, CFDIM);
    int strips_c = (NCOARSE / 16) * (HDIM / 64);              // 128
    wmma_gemm_f32<0, true><<<(strips_c + 7) / 8, 256, 0, stream>>>(ref_feats_c, W1, b1, o_hcref, NCOARSE, HDIM, CCDIM);
    wmma_gemm_f32<0, true><<<(strips_c + 7) / 8, 256, 0, stream>>>(src_feats_c, W1, b1, o_hcsrc, NCOARSE, HDIM, CCDIM);
  }

  // ---- raw scores = hf_ref @ hf_src^T into output buffer (L2-resident) ----
  {
    int strips = (NFINE / 16) * (NFINE / 64);                 // 16384
    wmma_gemm_f32<1, false><<<(strips + 7) / 8, 256, 0, stream>>>(hf_ref, hf_src, nullptr, o_scores, NFINE, NFINE, HDIM);
  }

  // ---- dual softmax (in place) + row/col argmax ----
  row_stats_kernel<<<NFINE, 256, 0, stream>>>(o_scores, rmax, rsum, NFINE);
  col_stats_kernel<<<NFINE / 256, 256, 0, stream>>>(o_scores, cmax, csum, NFINE, NFINE);
  dual_row_kernel<<<NFINE, 256, 0, stream>>>(o_scores, rmax, rsum, cmax, csum, conf_r, mi_r, NFINE);
  col_argmax_kernel<<<NFINE / 256, 256, 0, stream>>>(o_scores, conf_s, mi_s, NFINE, NFINE);

  // ---- compatibility (distances computed on the fly; matrices never stored) ----
  gather_pts_kernel<<<NFINE / 256, 256, 0, stream>>>(src_points, mi_r, q_r, NFINE);
  gather_pts_kernel<<<NFINE / 256, 256, 0, stream>>>(ref_points, mi_s, q_s, NFINE);
  compat_kernel<<<NFINE / 256, 256, 0, stream>>>(ref_points, q_r, o_compatr, NFINE);
  compat_kernel<<<NFINE / 256, 256, 0, stream>>>(src_points, q_s, o_compats, NFINE);

  // ---- seeding ----
  seed_kernel<<<1, 256, 0, stream>>>(o_compatr, conf_r, o_seedr, NFINE);
  seed_kernel<<<1, 256, 0, stream>>>(o_compats, conf_s, o_seeds, NFINE);

  // ---- KNN (only top-16 of the K_NN=33 is consumed; top-1 up; top-8 down) ----
  knn_kernel<SPOTK><<<NFINE / 256, 256, 0, stream>>>(ref_points, NFINE, ref_points, NFINE, ridx16, nullptr);
  knn_kernel<SPOTK><<<NFINE / 256, 256, 0, stream>>>(src_points, NFINE, src_points, NFINE, sidx16, nullptr);
  knn_kernel<1><<<NFINE / 256, 256, 0, stream>>>(ref_points, NFINE, ref_points_c, NCOARSE, nullptr, o_refup);
  knn_kernel<1><<<NFINE / 256, 256, 0, stream>>>(src_points, NFINE, src_points_c, NCOARSE, nullptr, o_srcup);
  knn_kernel<8><<<NCOARSE / 256, 256, 0, stream>>>(ref_points_c, NCOARSE, ref_points, NFINE, nullptr, o_refdn);
  knn_kernel<8><<<NCOARSE / 256, 256, 0, stream>>>(src_points_c, NCOARSE, src_points, NFINE, nullptr, o_srcdn);

  // ---- spots ----
  spot_kernel<<<(NFINE * SPOTK) / 256, 256, 0, stream>>>(mi_r, sidx16, o_spotr, NFINE);
  spot_kernel<<<(NFINE * SPOTK) / 256, 256, 0, stream>>>(mi_s, ridx16, o_spots, NFINE);
}